// MKMMDLoss_14654428414815
// MI455X (gfx1250) — compile-verified
//
#include <hip/hip_runtime.h>

typedef float v2f __attribute__((ext_vector_type(2)));
typedef float v8f __attribute__((ext_vector_type(8)));

#define NTOT  512   // concatenated rows (256 source + 256 target)
#define DIMK  512   // feature dimension
#define BHALF 256

__device__ __forceinline__ const float* row_ptr(const float* __restrict__ src,
                                                const float* __restrict__ tgt,
                                                int i) {
  return (i < BHALF) ? (src + (size_t)i * DIMK)
                     : (tgt + (size_t)(i - BHALF) * DIMK);
}

// ---------------------------------------------------------------------------
// Kernel 1: per-row squared norms sq[i] = sum_k T[i][k]^2
// ---------------------------------------------------------------------------
__global__ __launch_bounds__(256) void sq_kernel(const float* __restrict__ src,
                                                 const float* __restrict__ tgt,
                                                 float* __restrict__ sq) {
  int i = blockIdx.x * blockDim.x + threadIdx.x;
  if (i < NTOT) {
    const float* p = row_ptr(src, tgt, i);
    float s = 0.0f;
    for (int k = 0; k < DIMK; k += 4) {
      float4 v = *(const float4*)(p + k);
      s += v.x * v.x + v.y * v.y + v.z * v.z + v.w * v.w;
    }
    sq[i] = s;
  }
}

// ---------------------------------------------------------------------------
// Kernel 2: one wave32 per 16x16 output tile.
//   - Gram tile via V_WMMA_F32_16X16X4_F32 (fp32 matrix pipe, K-loop of 4)
//   - L1 tile via LDS-tiled VALU abs-diff accumulation
//   - fused epilogue: L2 = max(sq_i + sq_j - 2G, 0), store L2 & L1 matrices
// ---------------------------------------------------------------------------
__global__ __launch_bounds__(32) void tile_kernel(const float* __restrict__ src,
                                                  const float* __restrict__ tgt,
                                                  const float* __restrict__ sq,
                                                  float* __restrict__ L2m,
                                                  float* __restrict__ L1m) {
  const int ti   = blockIdx.y;          // row-tile  (0..31)
  const int tj   = blockIdx.x;          // col-tile  (0..31)
  const int lane = threadIdx.x;         // 0..31 (wave32)
  const int half = lane >> 4;           // 0: lanes 0-15, 1: lanes 16-31
  const int l    = lane & 15;

  // ISA 16x16x4 f32 operand layout:
  //   A (16x4, MxK): lane l<16 holds row M=l, (K=k0,k0+1); lanes 16-31 hold (K=k0+2,k0+3)
  //   B (4x16, KxN): lane l<16 holds col N=l, (K=k0,k0+1); lanes 16-31 hold (K=k0+2,k0+3)
  // -> both operands are contiguous float2 loads at column k0 + 2*half.
  const float* rowA = row_ptr(src, tgt, ti * 16 + l);
  const float* rowB = row_ptr(src, tgt, tj * 16 + l);

  v8f c = {0.f, 0.f, 0.f, 0.f, 0.f, 0.f, 0.f, 0.f};
#pragma unroll 4
  for (int k0 = 0; k0 < DIMK; k0 += 4) {
    v2f a = *(const v2f*)(rowA + k0 + 2 * half);
    v2f b = *(const v2f*)(rowB + k0 + 2 * half);
    // 8 args: (neg_a, A, neg_b, B, c_mod, C, reuse_a, reuse_b)
    c = __builtin_amdgcn_wmma_f32_16x16x4_f32(false, a, false, b,
                                              (short)0, c, false, false);
  }

  // ---- L1 distances for the same tile --------------------------------------
  // C/D layout: VGPR v -> (M = v + 8*half, N = l). Each lane owns 8 outputs.
  __shared__ float rb[16][68];          // 68-float stride: 16B-aligned rows, bank-skewed
  float acc[8];
#pragma unroll
  for (int v = 0; v < 8; ++v) acc[v] = 0.0f;

  const float* colRow = rowB;           // this lane's column row (N = l)

  for (int k0 = 0; k0 < DIMK; k0 += 64) {
    __syncthreads();
    // stage row-block chunk: 16 rows x 64 cols, 32 elems per lane, uniform row per step
    for (int idx = lane; idx < 16 * 64; idx += 32) {
      int r = idx >> 6, cc = idx & 63;
      rb[r][cc] = row_ptr(src, tgt, ti * 16 + r)[k0 + cc];
    }
    __syncthreads();

    for (int k = 0; k < 64; k += 4) {
      float4 cv = *(const float4*)(colRow + k0 + k);
#pragma unroll
      for (int v = 0; v < 8; ++v) {
        float4 rv = *(const float4*)(&rb[8 * half + v][k]);  // broadcast within half-wave
        acc[v] += fabsf(rv.x - cv.x) + fabsf(rv.y - cv.y) +
                  fabsf(rv.z - cv.z) + fabsf(rv.w - cv.w);
      }
    }
  }

  // ---- fused epilogue ------------------------------------------------------
  const int colN = tj * 16 + l;
  const float sqc = sq[colN];
#pragma unroll
  for (int v = 0; v < 8; ++v) {
    int r = ti * 16 + 8 * half + v;
    float g  = c[v];
    float d2 = fmaxf(sq[r] + sqc - 2.0f * g, 0.0f);
    L2m[(size_t)r * NTOT + colN] = d2;
    L1m[(size_t)r * NTOT + colN] = acc[v];
  }
}

// ---------------------------------------------------------------------------
// Kernel 3: deterministic global sums of L2/L1 -> bandwidths (single block)
//   bw = sum(dist) / (N^2 - N) / KERNEL_MUL^(KERNEL_NUM/2) = sum/(N^2-N)/100
// ---------------------------------------------------------------------------
__global__ __launch_bounds__(256) void bw_kernel(const float* __restrict__ L2m,
                                                 const float* __restrict__ L1m,
                                                 float* __restrict__ bw) {
  __shared__ float s2[256];
  __shared__ float s1[256];
  int tid = threadIdx.x;
  float a2 = 0.0f, a1 = 0.0f;
  for (int idx = tid; idx < NTOT * NTOT; idx += 256) {
    a2 += L2m[idx];
    a1 += L1m[idx];
  }
  s2[tid] = a2;
  s1[tid] = a1;
  __syncthreads();
  for (int off = 128; off > 0; off >>= 1) {
    if (tid < off) { s2[tid] += s2[tid + off]; s1[tid] += s1[tid + off]; }
    __syncthreads();
  }
  if (tid == 0) {
    const float denom = (float)(NTOT * NTOT - NTOT);
    bw[0] = s2[0] / denom * 0.01f;   // L2 base bandwidth
    bw[1] = s1[0] / denom * 0.01f;   // L1 base bandwidth
  }
}

// ---------------------------------------------------------------------------
// Kernel 4: signed multi-kernel sum, per-block deterministic tree reduction
// ---------------------------------------------------------------------------
__global__ __launch_bounds__(256) void ksum_kernel(const float* __restrict__ L2m,
                                                   const float* __restrict__ L1m,
                                                   const float* __restrict__ bw,
                                                   float* __restrict__ partials) {
  __shared__ float sm[256];
  const int tid = threadIdx.x;
  const float bw2 = bw[0], bw1 = bw[1];

  float inv2[5], inv1[5];
  float f = 1.0f;
#pragma unroll
  for (int p = 0; p < 5; ++p) {
    inv2[p] = -1.0f / (bw2 * f);
    inv1[p] = -1.0f / (bw1 * f);
    f *= 10.0f;
  }

  float local = 0.0f;
  const int tix = blockIdx.x * 256 + tid;       // 65536 threads
#pragma unroll
  for (int e = 0; e < 4; ++e) {                 // 4 strided elements / thread
    int idx = tix + e * 65536;                  // covers 512*512 = 262144
    int i = idx >> 9, j = idx & (NTOT - 1);
    float sgn = ((i < BHALF) == (j < BHALF)) ? 1.0f : -1.0f;
    float d2 = L2m[idx], d1 = L1m[idx];
    float kv = 0.0f;
#pragma unroll
    for (int p = 0; p < 5; ++p)
      kv += __expf(d2 * inv2[p]) + __expf(d1 * inv1[p]);
    local += sgn * kv;
  }

  sm[tid] = local;
  __syncthreads();
  for (int off = 128; off > 0; off >>= 1) {
    if (tid < off) sm[tid] += sm[tid + off];
    __syncthreads();
  }
  if (tid == 0) partials[blockIdx.x] = sm[0];
}

// ---------------------------------------------------------------------------
// Kernel 5: final reduction of 256 block partials -> mean
// ---------------------------------------------------------------------------
__global__ __launch_bounds__(256) void final_kernel(const float* __restrict__ partials,
                                                    float* __restrict__ out) {
  __shared__ float sm[256];
  int tid = threadIdx.x;
  sm[tid] = partials[tid];
  __syncthreads();
  for (int off = 128; off > 0; off >>= 1) {
    if (tid < off) sm[tid] += sm[tid + off];
    __syncthreads();
  }
  if (tid == 0) out[0] = sm[0] * (1.0f / 65536.0f);  // mean over 256x256 blocks
}

// ---------------------------------------------------------------------------
extern "C" void kernel_launch(void* const* d_in, const int* in_sizes, int n_in,
                              void* d_out, int out_size, void* d_ws, size_t ws_size,
                              hipStream_t stream) {
  (void)in_sizes; (void)n_in; (void)out_size; (void)ws_size;
  const float* src = (const float*)d_in[0];
  const float* tgt = (const float*)d_in[1];

  float* ws       = (float*)d_ws;
  float* sq       = ws;                          // 512
  float* L2m      = ws + 512;                    // 262144
  float* L1m      = L2m + NTOT * NTOT;           // 262144
  float* bw       = L1m + NTOT * NTOT;           // 2 (pad to 8)
  float* partials = bw + 8;                      // 256
  // total: ~2.1 MB of f32 workspace

  sq_kernel<<<2, 256, 0, stream>>>(src, tgt, sq);

  dim3 grid(NTOT / 16, NTOT / 16);               // 32 x 32 tiles, 1 wave each
  tile_kernel<<<grid, 32, 0, stream>>>(src, tgt, sq, L2m, L1m);

  bw_kernel<<<1, 256, 0, stream>>>(L2m, L1m, bw);
  ksum_kernel<<<256, 256, 0, stream>>>(L2m, L1m, bw, partials);
  final_kernel<<<1, 256, 0, stream>>>(partials, (float*)d_out);
}